// OrderedPredictor_54047868453214
// MI455X (gfx1250) — compile-verified
//
#include <hip/hip_runtime.h>

// ---------------------------------------------------------------------------
// Blocked forward substitution  (I - L) X = B  with 2048 RHS, DIM = 4096.
// fp32 throughout; matrix work done with V_WMMA_F32_16X16X4_F32 (CDNA5).
// Working set (L 64MB + R 32MB) fits in the 192MB L2 -> compute bound.
// ---------------------------------------------------------------------------

typedef __attribute__((ext_vector_type(2))) float v2f;
typedef __attribute__((ext_vector_type(8))) float v8f;

#define DIM    4096
#define BATCH  2048
#define NB     128          // panel size (K per GEMM step, trsm block rows)
#define TM     128          // GEMM workgroup tile M
#define TN     128          // GEMM workgroup tile N
#define APAD   4            // LDS padding (floats): stride 132 -> conflict-free frags
#define TRSM_COLS 256       // batch columns per trsm workgroup

// ---------------------------------------------------------------------------
// R[i][b] = noise[b][i], with row `target` forced to value.
// ---------------------------------------------------------------------------
__global__ void __launch_bounds__(256)
init_rhs(const float* __restrict__ noise, const float* __restrict__ value,
         const int* __restrict__ targetp, float* __restrict__ R) {
    const int target = *targetp;
    size_t idx = (size_t)blockIdx.x * blockDim.x + threadIdx.x;
    int i = (int)(idx / BATCH);
    int b = (int)(idx % BATCH);
    float v = (i == target) ? value[0] : noise[(size_t)b * DIM + i];
    R[idx] = v;   // R is [DIM][BATCH] row-major; idx == i*BATCH + b
}

// ---------------------------------------------------------------------------
// Diagonal-block forward substitution for rows [rowBase, rowBase+NB).
// Each thread owns one batch column; X block kept in LDS (dynamic j index).
// Row `target` does not mix (L row zeroed in the reference).
// ---------------------------------------------------------------------------
__global__ void __launch_bounds__(TRSM_COLS)
trsm_block(const float* __restrict__ L, float* __restrict__ R,
           int rowBase, const int* __restrict__ targetp) {
    extern __shared__ float smem[];
    float* Ls = smem;                       // NB x (NB+1)
    float* Xs = smem + NB * (NB + 1);       // NB x TRSM_COLS (column per thread)

    const int target = *targetp;
    const int tid = threadIdx.x;
    const int col = blockIdx.x * TRSM_COLS + tid;

    // Stage the diagonal L block (coalesced; j fastest).
    for (int t = 0; t < (NB * NB) / TRSM_COLS; ++t) {
        int linear = t * TRSM_COLS + tid;
        int r = linear / NB, j = linear % NB;
        int grow = rowBase + r;
        float a = (grow == target) ? 0.0f : L[(size_t)grow * DIM + rowBase + j];
        Ls[r * (NB + 1) + j] = a;
    }
    __syncthreads();

    // Sequential over rows; each thread only touches its own LDS column.
    for (int r = 0; r < NB; ++r) {
        float acc = R[(size_t)(rowBase + r) * BATCH + col];
        if (rowBase + r != target) {
            for (int j = 0; j < r; ++j)
                acc += Ls[r * (NB + 1) + j] * Xs[j * TRSM_COLS + tid];
        }
        Xs[r * TRSM_COLS + tid] = acc;
    }

    // Write solved block back (coalesced along batch).
    for (int r = 0; r < NB; ++r)
        R[(size_t)(rowBase + r) * BATCH + col] = Xs[r * TRSM_COLS + tid];
}

// ---------------------------------------------------------------------------
// Rank-NB update: R[rowStart.., :] += L[rowStart.., kBase..kBase+NB) * X[kBase rows]
// 256 threads = 8 waves; 128x128 C tile; wave w owns rows [16w,16w+16) x 128 cols.
// Fragments per CDNA5 ISA 7.12.2 fp32 layouts (h = lane>=16):
//   A 16x4 : {A[m][kk+2h], A[m][kk+2h+1]},  m = lane&15      (As row-major)
//   B 4x16 : {B[kk+2h][n], B[kk+2h+1][n]},  n = nt*16+lane&15 (Bs stored N-MAJOR
//            so the pair is one contiguous aligned ds_load_b64, no shuffles)
//   C 16x16: vgpr v -> (M = v + 8h, N = lane&15)
// ---------------------------------------------------------------------------
__global__ void __launch_bounds__(256)
gemm_update(const float* __restrict__ L, float* __restrict__ R,
            int rowStart, int kBase, const int* __restrict__ targetp) {
    extern __shared__ float smem[];
    float* As = smem;                        // TM x (NB+APAD), row-major (m, kk)
    float* Bs = smem + TM * (NB + APAD);     // TN x (NB+APAD), N-MAJOR (n, kk)

    const int target = *targetp;
    const int tid  = threadIdx.x;
    const int wave = tid >> 5;
    const int lane = tid & 31;
    const int half = lane >> 4;              // 0 or 1
    const int l15  = lane & 15;
    const int rowTile = rowStart + blockIdx.y * TM;
    const int colTile = blockIdx.x * TN;

    // Stage A panel of L (zero the intervention row). Coalesced: j fastest.
    for (int t = 0; t < (TM * NB) / 256; ++t) {
        int linear = t * 256 + tid;
        int r = linear / NB, j = linear % NB;
        int grow = rowTile + r;
        float a = (grow == target) ? 0.0f : L[(size_t)grow * DIM + kBase + j];
        As[r * (NB + APAD) + j] = a;
    }
    // Stage B panel = already-solved X rows of R, transposed into N-major LDS.
    // Global read stays coalesced (n fastest); LDS write eats a mild conflict.
    for (int t = 0; t < (NB * TN) / 256; ++t) {
        int linear = t * 256 + tid;
        int kk = linear / TN, n = linear % TN;
        Bs[n * (NB + APAD) + kk] = R[(size_t)(kBase + kk) * BATCH + colTile + n];
    }
    __syncthreads();

    const int m0 = wave * 16;

    // Initialize accumulators from R (read-modify-write; tile exclusively owned).
    v8f c[8];
#pragma unroll
    for (int n = 0; n < 8; ++n) {
        int gcol = colTile + n * 16 + l15;
#pragma unroll
        for (int v = 0; v < 8; ++v) {
            int grow = rowTile + m0 + v + half * 8;
            c[n][v] = R[(size_t)grow * BATCH + gcol];
        }
    }

    // K loop: one A fragment (b64) reused across the 8 N tiles; each B fragment
    // is a single aligned b64 LDS load thanks to the N-major Bs layout.
    const float* aRow = As + (m0 + l15) * (NB + APAD) + 2 * half;
    const float* bRow = Bs + l15 * (NB + APAD) + 2 * half;
#pragma unroll 4
    for (int kk = 0; kk < NB; kk += 4) {
        v2f a = *(const v2f*)(aRow + kk);
#pragma unroll
        for (int n = 0; n < 8; ++n) {
            v2f b = *(const v2f*)(bRow + n * 16 * (NB + APAD) + kk);
            c[n] = __builtin_amdgcn_wmma_f32_16x16x4_f32(
                /*neg_a=*/false, a, /*neg_b=*/false, b,
                /*c_mod=*/(short)0, c[n], /*reuse_a=*/false, /*reuse_b=*/false);
        }
    }

    // Store back.
#pragma unroll
    for (int n = 0; n < 8; ++n) {
        int gcol = colTile + n * 16 + l15;
#pragma unroll
        for (int v = 0; v < 8; ++v) {
            int grow = rowTile + m0 + v + half * 8;
            R[(size_t)grow * BATCH + gcol] = c[n][v];
        }
    }
}

// ---------------------------------------------------------------------------
// out[b][i] = R[i][b]
// ---------------------------------------------------------------------------
__global__ void __launch_bounds__(256)
writeout(const float* __restrict__ R, float* __restrict__ out) {
    size_t idx = (size_t)blockIdx.x * blockDim.x + threadIdx.x;
    int b = (int)(idx / DIM);
    int i = (int)(idx % DIM);
    out[idx] = R[(size_t)i * BATCH + b];
}

// ---------------------------------------------------------------------------
extern "C" void kernel_launch(void* const* d_in, const int* in_sizes, int n_in,
                              void* d_out, int out_size, void* d_ws, size_t ws_size,
                              hipStream_t stream) {
    const float* noise  = (const float*)d_in[0];  // [BATCH, DIM]
    const float* L      = (const float*)d_in[1];  // [DIM, DIM] strictly lower
    const float* value  = (const float*)d_in[2];  // [1]
    const int*   target = (const int*)d_in[3];    // scalar

    float* R = (float*)d_ws;                      // [DIM][BATCH] evolving RHS/X
    (void)in_sizes; (void)n_in; (void)out_size; (void)ws_size;

    const size_t nElem = (size_t)DIM * BATCH;
    const size_t trsmLds = (size_t)(NB * (NB + 1) + NB * TRSM_COLS) * sizeof(float);
    const size_t gemmLds = (size_t)((TM + TN) * (NB + APAD)) * sizeof(float);

    init_rhs<<<(unsigned)(nElem / 256), 256, 0, stream>>>(noise, value, target, R);

    const int nblk = DIM / NB;                    // 32 sequential panel steps
    for (int k = 0; k < nblk; ++k) {
        trsm_block<<<BATCH / TRSM_COLS, TRSM_COLS, trsmLds, stream>>>(
            L, R, k * NB, target);
        int mTiles = (DIM - (k + 1) * NB) / TM;
        if (mTiles > 0) {
            dim3 grid(BATCH / TN, mTiles);
            gemm_update<<<grid, 256, gemmLds, stream>>>(
                L, R, (k + 1) * NB, k * NB, target);
        }
    }

    writeout<<<(unsigned)(nElem / 256), 256, 0, stream>>>(R, (float*)d_out);
}